// CandidateExtractor_77429670412471
// MI455X (gfx1250) — compile-verified
//
#include <hip/hip_runtime.h>

// CandidateExtractor for MI455X (gfx1250, wave32).
// Bandwidth-bound: 2 passes over 256 MB heatmap (~22us floor @ 23.3 TB/s).
// Exact per-batch top-64 via 12-bit radix histogram + candidate pool + 64x argmax.
// 64x64 NMS distance Gram matrix computed with V_WMMA_F32_16X16X4_F32 (K=2 padded).

typedef float v2f __attribute__((ext_vector_type(2)));
typedef float v8f __attribute__((ext_vector_type(8)));

#define HDIM     1024
#define WDIM     1024
#define NPIX     (HDIM * WDIM)
#define NTHREADS 1024
#define NBUCKET  4096
#define CAP      2048
#define MAXK     64
#define NMS_R2   0.0025f   // 0.05^2

__device__ __forceinline__ unsigned monokey(float f) {
  // order-preserving float -> uint map (descending float == descending uint)
  unsigned u = __float_as_uint(f);
  return (u & 0x80000000u) ? ~u : (u | 0x80000000u);
}

__global__ __launch_bounds__(NTHREADS)
void cand_extract_kernel(const float* __restrict__ heat,
                         const int* __restrict__ ncp,
                         float* __restrict__ out) {
  __shared__ unsigned           hist[NBUCKET];        // 16 KB
  __shared__ unsigned long long cand[CAP];            // 16 KB
  __shared__ unsigned long long redK[NTHREADS];       //  8 KB
  __shared__ unsigned long long topKey[MAXK];
  __shared__ float cx[MAXK], cy[MAXK], sq[MAXK];
  __shared__ float d2[MAXK * MAXK];                   // 16 KB
  __shared__ int      s_cb;
  __shared__ unsigned s_cnt;

  const int tid = threadIdx.x;
  const int b   = blockIdx.x;

  const int nc = ncp[0];                 // num_candidates (16)
  int kk = nc * 4;                       // top-k pool size
  if (kk > MAXK) kk = MAXK;
  if (kk < 1)    kk = 1;
  int nkeep = nc;                        // outputs per batch
  if (nkeep > MAXK) nkeep = MAXK;

  // ---- Phase 0: clear LDS state ----
  for (int i = tid; i < NBUCKET; i += NTHREADS) hist[i] = 0u;
  if (tid == 0) s_cnt = 0u;
  __syncthreads();

  // ---- Phase 1: 12-bit key histogram over this batch's 1M pixels ----
  const float4* hp = (const float4*)(heat + (size_t)b * NPIX);
  for (int i = tid; i < NPIX / 4; i += NTHREADS) {
    float4 f = hp[i];
    atomicAdd(&hist[monokey(f.x) >> 20], 1u);
    atomicAdd(&hist[monokey(f.y) >> 20], 1u);
    atomicAdd(&hist[monokey(f.z) >> 20], 1u);
    atomicAdd(&hist[monokey(f.w) >> 20], 1u);
  }
  __syncthreads();

  // ---- Phase 2: cutoff bucket for rank kk ----
  if (tid == 0) {
    unsigned acc = 0; int cb = 0;
    for (int bb = NBUCKET - 1; bb >= 0; --bb) {
      acc += hist[bb];
      if (acc >= (unsigned)kk) { cb = bb; break; }
    }
    s_cb = cb;
  }
  __syncthreads();
  const unsigned cb = (unsigned)s_cb;

  // ---- Phase 3: collect candidates (bucket >= cutoff) into LDS pool ----
  for (int i = tid; i < NPIX / 4; i += NTHREADS) {
    float4 f = hp[i];
    float v[4] = {f.x, f.y, f.z, f.w};
#pragma unroll
    for (int c = 0; c < 4; ++c) {
      unsigned key = monokey(v[c]);
      if ((key >> 20) >= cb) {
        unsigned pos = atomicAdd(&s_cnt, 1u);
        if (pos < CAP) {
          unsigned idx = (unsigned)(4 * i + c);
          // high 32: value key; low 32: ~idx (ties -> lowest index first, like top_k)
          cand[pos] = ((unsigned long long)key << 32) |
                      (unsigned long long)(~idx);
        }
      }
    }
  }
  __syncthreads();
  int m = (int)s_cnt; if (m > CAP) m = CAP;

  // ---- Phase 4: kk rounds of block-wide argmax -> exact sorted top-kk ----
  for (int r = 0; r < kk; ++r) {
    unsigned long long lm = 0ull;
    for (int s = tid; s < m; s += NTHREADS) {
      unsigned long long v = cand[s];
      if (v > lm) lm = v;
    }
    redK[tid] = lm;
    __syncthreads();
    for (int off = NTHREADS / 2; off > 0; off >>= 1) {
      if (tid < off) {
        unsigned long long o = redK[tid + off];
        if (o > redK[tid]) redK[tid] = o;
      }
      __syncthreads();
    }
    unsigned long long w = redK[0];
    if (tid == 0) topKey[r] = w;
    for (int s = tid; s < m; s += NTHREADS)
      if (cand[s] == w) cand[s] = 0ull;   // retire winner
    __syncthreads();
  }

  // ---- Phase 5: decode normalized coords for the top-64 slots ----
  if (tid < MAXK) {
    float x = 0.f, y = 0.f;
    if (tid < kk && topKey[tid] != 0ull) {
      unsigned idx = ~((unsigned)(topKey[tid] & 0xFFFFFFFFull));
      unsigned iy = idx / WDIM, ix = idx % WDIM;
      x = (float)ix / (float)(WDIM - 1);
      y = (float)iy / (float)(HDIM - 1);
    }
    cx[tid] = x; cy[tid] = y; sq[tid] = x * x + y * y;
  }
  __syncthreads();

  // ---- Phase 6: 64x64 squared-distance matrix via f32 WMMA Gram tiles ----
  // G = C*C^T (C is 64x2, K padded 2->4). 16 tiles of 16x16, one per wave.
  {
    const int wv = tid >> 5;      // wave id 0..31
    const int ln = tid & 31;      // lane id
    if (wv < 16) {                // wave-uniform branch: EXEC stays all-ones
      const int ti = wv >> 2, tj = wv & 3;
      v2f a;  a.x  = 0.f; a.y  = 0.f;   // lanes 16-31 carry K=2,3 => zero pad
      v2f bb; bb.x = 0.f; bb.y = 0.f;
      if (ln < 16) {
        a.x  = cx[ti * 16 + ln];  a.y  = cy[ti * 16 + ln];   // A[M][K0..1]
        bb.x = cx[tj * 16 + ln];  bb.y = cy[tj * 16 + ln];   // B[K0..1][N]
      }
      v8f c = {0.f, 0.f, 0.f, 0.f, 0.f, 0.f, 0.f, 0.f};
      c = __builtin_amdgcn_wmma_f32_16x16x4_f32(
              false, a, false, bb, (short)0, c, false, false);
      const int rbase = ti * 16 + ((ln >> 4) << 3);   // lanes 16-31 -> rows M+8
      const int col   = tj * 16 + (ln & 15);
#pragma unroll
      for (int v = 0; v < 8; ++v) {
        const int row = rbase + v;
        d2[row * MAXK + col] = sq[row] + sq[col] - 2.0f * c[v];
      }
    }
  }
  __syncthreads();

  // ---- Phase 7: greedy sequential NMS (inherently serial), write output ----
  if (tid == 0) {
    float* ob = out + (size_t)b * (size_t)nkeep * 2;
    for (int i = 0; i < nkeep * 2; ++i) ob[i] = 0.f;   // zero-fill dropped slots
    unsigned long long selmask = 0ull;
    int cnt = 0;
    for (int i = 0; i < kk; ++i) {
      bool close = false;
      for (int j = 0; j < i; ++j) {
        if (((selmask >> j) & 1ull) && d2[i * MAXK + j] < NMS_R2) close = true;
      }
      const bool acc = (!close) && (cnt < nkeep);
      if (acc) {
        selmask |= (1ull << i);
        ob[cnt * 2 + 0] = cx[i];
        ob[cnt * 2 + 1] = cy[i];
        ++cnt;
      }
    }
  }
}

extern "C" void kernel_launch(void* const* d_in, const int* in_sizes, int n_in,
                              void* d_out, int out_size, void* d_ws, size_t ws_size,
                              hipStream_t stream) {
  const float* heat = (const float*)d_in[0];
  const int*   ncp  = (const int*)d_in[1];
  float*       out  = (float*)d_out;
  const int B = in_sizes[0] / NPIX;   // 64
  cand_extract_kernel<<<dim3(B), dim3(NTHREADS), 0, stream>>>(heat, ncp, out);
}